// TCRcube_AAidpos_35828617183363
// MI455X (gfx1250) — compile-verified
//
#include <hip/hip_runtime.h>

typedef __attribute__((ext_vector_type(2))) float v2f;
typedef __attribute__((ext_vector_type(8))) float v8f;

#define BATCH 32
#define IA 24
#define JJ 16
#define KK 48
#define EDIM 21
#define INNER 64

// ws layout (floats)
#define WS_A 0                      // 32*24*64
#define WS_B 49152                  // 32*24*64
#define WS_P 98304                  // 32*16*64
#define WS_M 131072                 // 32*48*64

// ---------------------------------------------------------------------------
// Kernel 1: embedding + positional encoding + linear projection to INNER=64.
// One 64-thread block per sequence row (3584 rows total).
// ---------------------------------------------------------------------------
__global__ __launch_bounds__(64)
void tcr_proj_kernel(const int* __restrict__ tokA, const int* __restrict__ tokB,
                     const int* __restrict__ tokP, const int* __restrict__ tokM,
                     const float* __restrict__ emb, const float* __restrict__ pos,
                     const float* __restrict__ Wa, const float* __restrict__ ba,
                     const float* __restrict__ Wb, const float* __restrict__ bb,
                     const float* __restrict__ Wp, const float* __restrict__ bp,
                     const float* __restrict__ Wm, const float* __restrict__ bm,
                     float* __restrict__ ws) {
    __shared__ float ybuf[EDIM];
    const int r = blockIdx.x;            // 0 .. 32*112-1
    const int b = r / 112;
    const int off = r % 112;

    const int* tok; const float* W; const float* bias; float* dst; int n; int seqlen;
    if (off < 24)      { tok = tokA; W = Wa; bias = ba; n = off;      seqlen = IA; dst = ws + WS_A + (b*IA + n)*INNER; }
    else if (off < 48) { tok = tokB; W = Wb; bias = bb; n = off - 24; seqlen = IA; dst = ws + WS_B + (b*IA + n)*INNER; }
    else if (off < 64) { tok = tokP; W = Wp; bias = bp; n = off - 48; seqlen = JJ; dst = ws + WS_P + (b*JJ + n)*INNER; }
    else               { tok = tokM; W = Wm; bias = bm; n = off - 64; seqlen = KK; dst = ws + WS_M + (b*KK + n)*INNER; }

    const int t = tok[b*seqlen + n];     // tokens are (B, N, 1) int32
    const int d = threadIdx.x;
    if (d < EDIM) {
        float ev = emb[t*EDIM + d];
        // reference: y = where(e == 0, 0, e + pos)
        ybuf[d] = (ev == 0.0f) ? 0.0f : (ev + pos[n*EDIM + d]);
    }
    __syncthreads();

    const int o = threadIdx.x;           // 0..63
    float s = bias[o];
    #pragma unroll
    for (int dd = 0; dd < EDIM; ++dd) s += ybuf[dd] * W[o*EDIM + dd];
    dst[o] = s;
}

// ---------------------------------------------------------------------------
// Kernel 2: fused cube / layernorm / relu / dot / masked-reduce energy.
// One 128-thread block (4 waves) per (batch, half). 72 WMMA 16x16 tiles per
// block: S1 = (x*y) @ z, S2 = (x*y)^2 @ z^2 via V_WMMA_F32_16X16X4_F32.
// ---------------------------------------------------------------------------
__global__ __launch_bounds__(128)
void tcr_energy_kernel(const float* __restrict__ ws,
                       const float* __restrict__ maskA, const float* __restrict__ maskB,
                       const float* __restrict__ maskP, const float* __restrict__ maskM,
                       const float* __restrict__ lens,
                       const float* __restrict__ anorm_g, const float* __restrict__ anorm_b,
                       const float* __restrict__ bnorm_g, const float* __restrict__ bnorm_b,
                       const float* __restrict__ aout_w, const float* __restrict__ aout_b,
                       const float* __restrict__ bout_w, const float* __restrict__ bout_b,
                       float* __restrict__ out) {
    __shared__ float xlds[IA*INNER];      // cdr_s  (i, e)
    __shared__ float ylds[JJ*INNER];      // pep_s  (j, e)
    __shared__ float zlds[INNER*KK];      // mhc_s  transposed (e, k)
    __shared__ float z2lds[INNER*KK];     // mhc_s^2 transposed
    __shared__ float glds[INNER], bnlds[INNER], wlds[INNER];
    __shared__ float mcl[IA], mpl[JJ], mml[KK];
    __shared__ float redbuf[4];

    const int blk  = blockIdx.x;          // 0..63
    const int b    = blk >> 1;
    const int half = blk & 1;
    const int tid  = threadIdx.x;

    const float* cdr = ws + (half ? WS_B : WS_A) + b*IA*INNER;
    const float* pep = ws + WS_P + b*JJ*INNER;
    const float* mhc = ws + WS_M + b*KK*INNER;
    const float* g   = half ? bnorm_g : anorm_g;
    const float* bn  = half ? bnorm_b : anorm_b;
    const float* w   = half ? bout_w  : aout_w;
    const float  outb = half ? bout_b[0] : aout_b[0];
    const float* mc  = half ? maskB : maskA;

    for (int idx = tid; idx < IA*INNER; idx += 128) xlds[idx] = cdr[idx];
    for (int idx = tid; idx < JJ*INNER; idx += 128) ylds[idx] = pep[idx];
    for (int idx = tid; idx < KK*INNER; idx += 128) {
        int k = idx >> 6, e = idx & 63;
        float v = mhc[idx];
        zlds[e*KK + k]  = v;
        z2lds[e*KK + k] = v * v;
    }
    if (tid < INNER) { glds[tid] = g[tid]; bnlds[tid] = bn[tid]; wlds[tid] = w[tid]; }
    if (tid < IA) mcl[tid] = mc[b*IA + tid];
    if (tid < JJ) mpl[tid] = maskP[b*JJ + tid];
    if (tid < KK) mml[tid] = maskM[b*KK + tid];
    __syncthreads();

    const int wave  = tid >> 5;
    const int lane  = tid & 31;
    const int halfw = lane >> 4;          // upper/lower 16 lanes
    const int lm    = lane & 15;

    float msum = 0.0f;

    for (int t = wave; t < IA*3; t += 4) {        // 72 tiles / 4 waves
        const int i  = t / 3;                     // cdr index (tile row group)
        const int nt = t % 3;                     // k-tile
        const int k  = nt*16 + lm;                // this lane's output column
        const float* xr = &xlds[i*INNER];

        v8f s1 = {0,0,0,0,0,0,0,0};
        v8f s2 = {0,0,0,0,0,0,0,0};

        // A fragment: 16x4 f32, M = lane&15 (row j), VGPR0 holds K=e0+2*halfw,
        // VGPR1 holds K=e0+2*halfw+1.  B fragment mirrors with N = lane&15.
        const int jA = lm;
        #pragma unroll
        for (int e0 = 0; e0 < INNER; e0 += 4) {
            const int ea = e0 + 2*halfw;
            v2f a, asq, bf, bsq;
            float x0 = xr[ea],            x1 = xr[ea + 1];
            float y0 = ylds[jA*INNER+ea], y1 = ylds[jA*INNER+ea+1];
            a[0]   = x0 * y0;             a[1]   = x1 * y1;
            asq[0] = a[0] * a[0];         asq[1] = a[1] * a[1];
            bf[0]  = zlds[ea*KK + k];     bf[1]  = zlds[(ea+1)*KK + k];
            bsq[0] = z2lds[ea*KK + k];    bsq[1] = z2lds[(ea+1)*KK + k];
            s1 = __builtin_amdgcn_wmma_f32_16x16x4_f32(false, a,   false, bf,  (short)0, s1, false, false);
            s2 = __builtin_amdgcn_wmma_f32_16x16x4_f32(false, asq, false, bsq, (short)0, s2, false, false);
        }

        const float mi = mcl[i];
        const float mk = mml[k];
        #pragma unroll
        for (int v = 0; v < 8; ++v) {
            const int j = v + 8*halfw;            // C/D layout: M = v + 8*(lane>=16)
            float mu  = s1[v] * (1.0f/64.0f);
            float var = s2[v] * (1.0f/64.0f) - mu*mu;
            float rs  = rsqrtf(var + 1e-5f);
            float acc = 0.0f;
            const float* yr = &ylds[j*INNER];
            #pragma unroll 8
            for (int e = 0; e < INNER; ++e) {
                float c  = xr[e] * yr[e] * zlds[e*KK + k];
                float tt = (c - mu) * rs * glds[e] + bnlds[e];
                acc += fmaxf(tt, 0.0f) * wlds[e];
            }
            msum += (acc + outb) * mi * mpl[j] * mk;
        }
    }

    // wave32 reduction, then cross-wave via LDS
    for (int o = 16; o > 0; o >>= 1) msum += __shfl_down(msum, o, 32);
    if (lane == 0) redbuf[wave] = msum;
    __syncthreads();
    if (tid == 0) {
        float tot = redbuf[0] + redbuf[1] + redbuf[2] + redbuf[3];
        float lc = lens[b*4 + (half ? 1 : 0)];
        float lp = lens[b*4 + 2];
        float lmm = lens[b*4 + 3];
        atomicAdd(&out[b], tot / (lc * lp * lmm));
    }
}

extern "C" void kernel_launch(void* const* d_in, const int* in_sizes, int n_in,
                              void* d_out, int out_size, void* d_ws, size_t ws_size,
                              hipStream_t stream) {
    const int*   CDR3a   = (const int*)d_in[0];
    const int*   CDR3b   = (const int*)d_in[1];
    const int*   peptide = (const int*)d_in[2];
    const int*   mhc     = (const int*)d_in[3];
    const float* maskA   = (const float*)d_in[4];
    const float* maskB   = (const float*)d_in[5];
    const float* maskP   = (const float*)d_in[6];
    const float* maskM   = (const float*)d_in[7];
    const float* lens    = (const float*)d_in[8];
    const float* emb     = (const float*)d_in[9];
    const float* pos     = (const float*)d_in[10];
    const float* Wa      = (const float*)d_in[11];
    const float* ba      = (const float*)d_in[12];
    const float* Wb      = (const float*)d_in[13];
    const float* bb      = (const float*)d_in[14];
    const float* Wp      = (const float*)d_in[15];
    const float* bp      = (const float*)d_in[16];
    const float* Wm      = (const float*)d_in[17];
    const float* bm      = (const float*)d_in[18];
    const float* anorm_g = (const float*)d_in[19];
    const float* anorm_b = (const float*)d_in[20];
    const float* bnorm_g = (const float*)d_in[21];
    const float* bnorm_b = (const float*)d_in[22];
    const float* aout_w  = (const float*)d_in[23];
    const float* aout_b  = (const float*)d_in[24];
    const float* bout_w  = (const float*)d_in[25];
    const float* bout_b  = (const float*)d_in[26];

    float* ws  = (float*)d_ws;
    float* out = (float*)d_out;

    hipMemsetAsync(d_out, 0, out_size * sizeof(float), stream);

    tcr_proj_kernel<<<BATCH*112, 64, 0, stream>>>(
        CDR3a, CDR3b, peptide, mhc, emb, pos,
        Wa, ba, Wb, bb, Wp, bp, Wm, bm, ws);

    tcr_energy_kernel<<<BATCH*2, 128, 0, stream>>>(
        ws, maskA, maskB, maskP, maskM, lens,
        anorm_g, anorm_b, bnorm_g, bnorm_b,
        aout_w, aout_b, bout_w, bout_b, out);
}